// _HashGridLoRA_33397665693996
// MI455X (gfx1250) — compile-verified
//
#include <hip/hip_runtime.h>
#include <hip/hip_bf16.h>

typedef __attribute__((ext_vector_type(2))) float v2f;
typedef __attribute__((ext_vector_type(4))) float v4f;
typedef __attribute__((ext_vector_type(8))) float v8f;

#define HASH_P1 2654435761u
#define HASH_P2 805459861u
#define HASH_MASK 524287u   // HASHMAP_SIZE - 1 (2^19)
#define LR_N 2048
#define LR_K 64

// ---------------------------------------------------------------------------
// Phase 1: LR = L(1024x64) @ R(64x2048) in fp32 via V_WMMA_F32_16X16X4_F32.
// One wave32 computes one 16x16 output tile; K=64 -> 16 WMMA steps.
// ---------------------------------------------------------------------------
__global__ void __launch_bounds__(128)
lr_gemm_wmma(const float* __restrict__ L, const float* __restrict__ R,
             float* __restrict__ LR) {
  const int lane = threadIdx.x;                 // 0..31
  const int tile = blockIdx.x * 4 + threadIdx.y;
  const int TILES_N = LR_N / 16;                // 128
  const int tm = tile / TILES_N;
  const int tn = tile % TILES_N;

  // fp32 WMMA A(16x4)/B(4x16) lane layout: m/n = lane&15, kbase = 2*(lane>>4)
  const int mn = lane & 15;
  const int kb = (lane >> 4) << 1;

  const float* __restrict__ Lrow = L + (tm * 16 + mn) * LR_K;
  const float* __restrict__ Bcol = R + (tn * 16 + mn);

  v8f acc = {};
#pragma unroll
  for (int k0 = 0; k0 < LR_K; k0 += 4) {
    v2f a, b;
    a.x = Lrow[k0 + kb + 0];
    a.y = Lrow[k0 + kb + 1];
    b.x = Bcol[(k0 + kb + 0) * LR_N];
    b.y = Bcol[(k0 + kb + 1) * LR_N];
    acc = __builtin_amdgcn_wmma_f32_16x16x4_f32(
        /*neg_a=*/false, a, /*neg_b=*/false, b,
        /*c_mod=*/(short)0, acc, /*reuse_a=*/false, /*reuse_b=*/false);
  }

  // D layout: VGPR r -> M = r + 8*(lane>>4), N = lane&15
  const int mbase = (lane >> 4) * 8;
#pragma unroll
  for (int r = 0; r < 8; ++r) {
    LR[(tm * 16 + mbase + r) * LR_N + tn * 16 + mn] = acc[r];
  }
}

// ---------------------------------------------------------------------------
// Phase 2: hash-grid trilinear interpolation. One point per lane.
// Table (8 MB) is L2-resident; keep gathers RT, stream x/out with NT hints.
// ---------------------------------------------------------------------------
__global__ void __launch_bounds__(256)
hashgrid_gather(const float* __restrict__ x, const float* __restrict__ LR,
                float* __restrict__ out, int n_points) {
  const int i = blockIdx.x * blockDim.x + threadIdx.x;
  if (i >= n_points) return;

  const float px = __builtin_nontemporal_load(x + 3 * i + 0);
  const float py = __builtin_nontemporal_load(x + 3 * i + 1);
  const float pz = __builtin_nontemporal_load(x + 3 * i + 2);

  const float sx = px * 128.0f, sy = py * 128.0f, sz = pz * 128.0f;
  const float fx0 = floorf(sx), fy0 = floorf(sy), fz0 = floorf(sz);
  const float xf = sx - fx0, yf = sy - fy0, zf = sz - fz0;
  const unsigned ix = (unsigned)fx0, iy = (unsigned)fy0, iz = (unsigned)fz0;

  // Per-axis hash contributions and weights; corner bit d: 0 -> base/(1-f), 1 -> +1/f
  const unsigned hx[2] = { ix,              ix + 1u };             // prime = 1
  const unsigned hy[2] = { iy * HASH_P1,   (iy + 1u) * HASH_P1 };
  const unsigned hz[2] = { iz * HASH_P2,   (iz + 1u) * HASH_P2 };
  const float wx[2] = { 1.0f - xf, xf };
  const float wy[2] = { 1.0f - yf, yf };
  const float wz[2] = { 1.0f - zf, zf };

  const v4f* __restrict__ tab = (const v4f*)LR;
  v4f acc = {0.0f, 0.0f, 0.0f, 0.0f};

#pragma unroll
  for (int c = 0; c < 8; ++c) {
    const int bx = c & 1, by = (c >> 1) & 1, bz = (c >> 2) & 1;
    const unsigned h = (hx[bx] ^ hy[by] ^ hz[bz]) & HASH_MASK;
    const float w = wx[bx] * wy[by] * wz[bz];
    const v4f f = tab[h];
    acc.x = fmaf(w, f.x, acc.x);
    acc.y = fmaf(w, f.y, acc.y);
    acc.z = fmaf(w, f.z, acc.z);
    acc.w = fmaf(w, f.w, acc.w);
  }

  __builtin_nontemporal_store(acc, (v4f*)out + i);
}

extern "C" void kernel_launch(void* const* d_in, const int* in_sizes, int n_in,
                              void* d_out, int out_size, void* d_ws, size_t ws_size,
                              hipStream_t stream) {
  const float* x = (const float*)d_in[0];   // (N_POINTS, 3)
  const float* L = (const float*)d_in[1];   // (1024, 64)
  const float* R = (const float*)d_in[2];   // (64, 2048)
  float* out = (float*)d_out;               // (N_POINTS, 4)
  float* LR = (float*)d_ws;                 // 1024*2048 floats = 8 MB scratch

  // Phase 1: 64x128 = 8192 tiles of 16x16; 4 waves/block -> 2048 blocks
  dim3 gemm_block(32, 4);
  lr_gemm_wmma<<<2048, gemm_block, 0, stream>>>(L, R, LR);

  // Phase 2: one thread per point
  const int n_points = in_sizes[0] / 3;
  const int threads = 256;
  const int blocks = (n_points + threads - 1) / threads;
  hashgrid_gather<<<blocks, threads, 0, stream>>>(x, LR, out, n_points);
}